// Kernel_estimation_3667902071333
// MI455X (gfx1250) — compile-verified
//
#include <hip/hip_runtime.h>

// Problem constants (fixed by the reference)
#define KS   5
#define TAPS 25          // KS*KS
#define PAD  2
#define NB   4
#define CB   8
#define HB   256
#define WB   256
#define TILE 16
#define XT   (TILE + 2 * PAD)   // 20

// Types matching the clang-23 async-to-LDS builtin signatures:
//   __builtin_amdgcn_global_load_async_to_lds_b128(v4i AS1*, v4i AS3*, imm, imm)
//   __builtin_amdgcn_global_load_async_to_lds_b32 (int AS1*, int AS3*, imm, imm)
typedef int v4i __attribute__((ext_vector_type(4)));
typedef __attribute__((address_space(1))) v4i GV4;
typedef __attribute__((address_space(3))) v4i LV4;
typedef __attribute__((address_space(1))) int GI32;
typedef __attribute__((address_space(3))) int LI32;

__global__ __launch_bounds__(256) void kpn_dynconv_kernel(
    const float* __restrict__ x,      // (N, C, H, W)
    const float* __restrict__ kern,   // (N, C*25, H, W) == flat (N,C,H,W,5,5)
    float* __restrict__ out)          // (N, C, H, W)
{
    // Weights tile: 16 rows x (16 px * 25 taps) floats = 25,600 B, layout r*400 + c*25 + t.
    __shared__ __attribute__((aligned(16))) float wtile[TILE * TILE * TAPS];
    // x tile with 2-halo: 20x20 floats.
    __shared__ __attribute__((aligned(16))) float xtile[XT * XT];

    const int tid = threadIdx.x;
    __builtin_assume(tid < 256);

    const int tpp     = (HB / TILE) * (WB / TILE);          // tiles per plane = 256
    const int plane   = blockIdx.x / tpp;                   // n*C + c
    const int tile_id = blockIdx.x % tpp;
    const int th0     = (tile_id / (WB / TILE)) * TILE;     // tile origin h
    const int tw0     = (tile_id % (WB / TILE)) * TILE;     // tile origin w

    const float* kbase = kern + (size_t)plane * HB * WB * TAPS;
    const float* xbase = x    + (size_t)plane * HB * WB;

    // ---- Stage per-pixel weights via the CDNA5 async memory->LDS DMA path -------------------
    // Tile row r: ((th0+r)*W + tw0)*25 floats base, 1600 B contiguous, 16 B aligned.
    // 1600 b128 chunks per block: fully coalesced streaming of the dominant 209.7 MB tensor.
#pragma unroll
    for (int k = 0; k < 6; ++k) {
        const int idx = tid + k * 256;     // 0..1535
        const int r   = idx / 100;         // tile row
        const int c4  = idx % 100;         // float4 index within row
        const float* gsrc = kbase + ((size_t)(th0 + r) * WB + tw0) * TAPS + c4 * 4;
        __builtin_amdgcn_global_load_async_to_lds_b128(
            (GV4*)gsrc, (LV4*)&wtile[r * 400 + c4 * 4], 0, 0);
    }
    if (tid < 64) {                        // tail: chunks 1536..1599 (rows 15.36.. -> r=15)
        const int idx = 1536 + tid;
        const int r   = idx / 100;
        const int c4  = idx % 100;
        const float* gsrc = kbase + ((size_t)(th0 + r) * WB + tw0) * TAPS + c4 * 4;
        __builtin_amdgcn_global_load_async_to_lds_b128(
            (GV4*)gsrc, (LV4*)&wtile[r * 400 + c4 * 4], 0, 0);
    }
    // ---- Stage x halo tile (20x20 = 400 elems) with edge clamping (replication pad) ---------
    {
        const int idx = tid;               // pass 1: 0..255
        const int xr = idx / XT, xc = idx % XT;
        int gh = th0 + xr - PAD; gh = gh < 0 ? 0 : (gh > HB - 1 ? HB - 1 : gh);
        int gw = tw0 + xc - PAD; gw = gw < 0 ? 0 : (gw > WB - 1 ? WB - 1 : gw);
        __builtin_amdgcn_global_load_async_to_lds_b32(
            (GI32*)(xbase + gh * WB + gw), (LI32*)&xtile[idx], 0, 0);
    }
    if (tid < XT * XT - 256) {             // pass 2: 256..399
        const int idx = 256 + tid;
        const int xr = idx / XT, xc = idx % XT;
        int gh = th0 + xr - PAD; gh = gh < 0 ? 0 : (gh > HB - 1 ? HB - 1 : gh);
        int gw = tw0 + xc - PAD; gw = gw < 0 ? 0 : (gw > WB - 1 ? WB - 1 : gw);
        __builtin_amdgcn_global_load_async_to_lds_b32(
            (GI32*)(xbase + gh * WB + gw), (LI32*)&xtile[idx], 0, 0);
    }
    __builtin_amdgcn_s_wait_asynccnt(0);
    __syncthreads();

    // ---- Compute: one pixel per thread ------------------------------------------------------
    const int ty = tid >> 4;
    const int tx = tid & 15;
    // LDS reads: wtile lane stride 25 words (25 coprime 64 banks) and xtile rows -> conflict-free.
    const float* wp = &wtile[ty * (TILE * TAPS) + tx * TAPS];
    float acc = 0.0f;
#pragma unroll
    for (int i = 0; i < KS; ++i) {
#pragma unroll
        for (int j = 0; j < KS; ++j) {
            acc = fmaf(wp[i * KS + j], xtile[(ty + i) * XT + (tx + j)], acc);
        }
    }
    acc = acc > 0.0f ? acc : 0.2f * acc;  // leaky_relu, negative_slope = 0.2

    out[(size_t)plane * HB * WB + (size_t)(th0 + ty) * WB + (tw0 + tx)] = acc;
}

extern "C" void kernel_launch(void* const* d_in, const int* in_sizes, int n_in,
                              void* d_out, int out_size, void* d_ws, size_t ws_size,
                              hipStream_t stream) {
    const float* x    = (const float*)d_in[0];   // (4, 8, 256, 256) f32
    const float* kern = (const float*)d_in[1];   // (4, 200, 256, 256) f32
    float* out        = (float*)d_out;           // (4, 8, 256, 256) f32
    (void)in_sizes; (void)n_in; (void)out_size; (void)d_ws; (void)ws_size;

    const int blocks = NB * CB * (HB / TILE) * (WB / TILE);  // 8192
    hipLaunchKernelGGL(kpn_dynconv_kernel, dim3(blocks), dim3(256), 0, stream,
                       x, kern, out);
}